// SurpriseGatedSSD_29815662969090
// MI455X (gfx1250) — compile-verified
//
#include <hip/hip_runtime.h>
#include <math.h>

#define HNUM 16
#define DDIM 64
#define SDIM 64
#define BSZ 2
#define LSEQ 4096
#define TCHUNK 64
#define NCHUNK (LSEQ / TCHUNK)
#define TSTRIDE (HNUM * DDIM) /* 1024 floats between consecutive timesteps */
#define PAD 68                /* LDS row stride: 16B-aligned, bank-conflict-free */

typedef float v2f __attribute__((ext_vector_type(2)));
typedef float v8f __attribute__((ext_vector_type(8)));

static __device__ __forceinline__ v8f wmma_f32(v2f a, v2f b, v8f c) {
  return __builtin_amdgcn_wmma_f32_16x16x4_f32(false, a, false, b, (short)0, c,
                                               false, false);
}

// ---------------------------------------------------------------------------
// Pass 1: fully parallel over (b, h, chunk) = 2048 blocks.
// Computes gates, Y_intra = (G.M)X -> d_out, S_c = (W.B)^T X -> ws,
// e^{L_t} -> ws, D_c = e^{L63} -> ws.
// ---------------------------------------------------------------------------
#define P1_SMEM_FLOATS (4 * 64 * PAD + 5 * 64)

__global__ __launch_bounds__(256) void sgr_pass1(
    const float* __restrict__ X, const float* __restrict__ A,
    const float* __restrict__ Bm, const float* __restrict__ Cm,
    const float* __restrict__ l2ab, const float* __restrict__ l2b,
    const float* __restrict__ sema, float* __restrict__ Y,
    float* __restrict__ wsS, float* __restrict__ wsEL,
    float* __restrict__ wsDc) {
  extern __shared__ float smem[];
  float* sX = smem;
  float* sB = sX + 64 * PAD;
  float* sC = sB + 64 * PAD;
  float* sM = sC + 64 * PAD;
  float* sLa = sM + 64 * PAD;
  float* sLc = sLa + 64;
  float* sOma = sLc + 64;
  float* sEL = sOma + 64;
  float* sW = sEL + 64;

  const int tid = threadIdx.x;
  const int c = blockIdx.x & (NCHUNK - 1);
  const int bh = blockIdx.x >> 6;
  const int head = bh & (HNUM - 1);
  const int b = bh >> 4;
  const int t0 = c * TCHUNK;

  float ab = l2ab[head];
  ab = fminf(fmaxf(ab, -3.32f), -0.015f);
  const float alpha_base = 1.0f - exp2f(ab);
  float btv = l2b[head];
  btv = fminf(fmaxf(btv, -2.0f), 2.0f);
  const float beta = exp2f(btv);
  const float inv_ema = 1.0f / (sema[head] + 1e-6f);

  const int wave = tid >> 5;
  const int lane = tid & 31;
  const int m = lane & 15;
  const int half = lane >> 4;
  const int kb = half * 2;
  const size_t baseBL = (size_t)b * LSEQ;

  // ---- stage X, B, C chunk into LDS (float4 global loads, padded rows) ----
  {
    const size_t gbase = (baseBL + (size_t)t0) * TSTRIDE + (size_t)head * DDIM;
#pragma unroll
    for (int rep = 0; rep < 4; ++rep) {
      const int idx4 = rep * 256 + tid;
      const int r = idx4 >> 4;
      const int j = (idx4 & 15) * 4;
      const size_t g = gbase + (size_t)r * TSTRIDE + (size_t)j;
      *(float4*)&sX[r * PAD + j] = *(const float4*)&X[g];
      *(float4*)&sB[r * PAD + j] = *(const float4*)&Bm[g];
      *(float4*)&sC[r * PAD + j] = *(const float4*)&Cm[g];
    }
  }
  __syncthreads();

  // ---- per-timestep alpha_t (input-only; h_red carry is dead code) ----
  {
    const int t = tid >> 2, q = tid & 3;
    float xp = 0.0f, bp = 0.0f;
    const int base = t * PAD + q * 16;
#pragma unroll
    for (int j2 = 0; j2 < 16; ++j2) {
      const float xv = sX[base + j2];
      const float bv = sB[base + j2];
      xp += xv;
      bp += bv * bv;
    }
    xp += __shfl_xor(xp, 1, 4);
    xp += __shfl_xor(xp, 2, 4);
    bp += __shfl_xor(bp, 1, 4);
    bp += __shfl_xor(bp, 2, 4);
    if (q == 0) {
      const float a_log = A[(baseBL + (size_t)(t0 + t)) * HNUM + head];
      const float ad = __expf(a_log);
      const float one_m_ad = 1.0f - ad;
      const float xmean = xp * (1.0f / 64.0f);
      const float pred_err =
          one_m_ad * one_m_ad * xmean * xmean * (bp * (1.0f / 64.0f));
      float boost = fmaxf(tanhf(beta * pred_err * inv_ema), 0.0f);
      float alpha = alpha_base + (1.0f - alpha_base) * boost;
      alpha = fminf(fmaxf(alpha, 0.01f), 0.999f);
      sLa[t] = __logf(alpha);
      sOma[t] = 1.0f - alpha;
    }
  }
  __syncthreads();
  if (tid == 0) {
    float acc = 0.0f;
    for (int i = 0; i < 64; ++i) {
      acc += sLa[i];
      sLc[i] = acc;
    }
  }
  __syncthreads();
  if (tid < 64) {
    const float Lt = sLc[tid];
    const float L63 = sLc[63];
    sEL[tid] = __expf(Lt);
    sW[tid] = __expf(L63 - Lt) * sOma[tid];
    wsEL[(size_t)bh * LSEQ + (size_t)(t0 + tid)] = sEL[tid];
    if (tid == 63) wsDc[bh * NCHUNK + c] = sEL[63];
  }
  __syncthreads();

  // ---- matmul 1: M = C . B^T ----
#pragma unroll
  for (int p = 0; p < 2; ++p) {
    const int tile = wave * 2 + p, tm = tile >> 2, tn = tile & 3;
    const int arow = tm * 16 + m, brow = tn * 16 + m;
    v8f acc = {};
#pragma unroll
    for (int k0 = 0; k0 < 64; k0 += 4) {
      const int kk = k0 + kb;
      v2f av, bv;
      av[0] = sC[arow * PAD + kk];
      av[1] = sC[arow * PAD + kk + 1];
      bv[0] = sB[brow * PAD + kk];
      bv[1] = sB[brow * PAD + kk + 1];
      acc = wmma_f32(av, bv, acc);
    }
#pragma unroll
    for (int r = 0; r < 8; ++r) {
      const int row = tm * 16 + r + half * 8;
      sM[row * PAD + tn * 16 + m] = acc[r];
    }
  }
  __syncthreads();

  // ---- causal gate in log space ----
  {
    const int t = tid >> 2, q = tid & 3;
    const float Lt = sLc[t];
#pragma unroll
    for (int j2 = 0; j2 < 16; ++j2) {
      const int i = q * 16 + j2;
      float v = 0.0f;
      if (i <= t) v = sM[t * PAD + i] * __expf(Lt - sLc[i]) * sOma[i];
      sM[t * PAD + i] = v;
    }
  }
  __syncthreads();

  // ---- matmul 2: Y_intra = (G.M) . X  -> d_out ----
#pragma unroll
  for (int p = 0; p < 2; ++p) {
    const int tile = wave * 2 + p, tm = tile >> 2, tn = tile & 3;
    const int arow = tm * 16 + m, ncol = tn * 16 + m;
    v8f acc = {};
#pragma unroll
    for (int k0 = 0; k0 < 64; k0 += 4) {
      const int kk = k0 + kb;
      v2f av, bv;
      av[0] = sM[arow * PAD + kk];
      av[1] = sM[arow * PAD + kk + 1];
      bv[0] = sX[kk * PAD + ncol];
      bv[1] = sX[(kk + 1) * PAD + ncol];
      acc = wmma_f32(av, bv, acc);
    }
#pragma unroll
    for (int r = 0; r < 8; ++r) {
      const int trow = tm * 16 + r + half * 8;
      Y[(baseBL + (size_t)(t0 + trow)) * TSTRIDE + (size_t)head * DDIM + ncol] =
          acc[r];
    }
  }

  // ---- matmul 3: S_c = (W.B)^T . X  -> ws (no barrier needed: no LDS writes) ----
#pragma unroll
  for (int p = 0; p < 2; ++p) {
    const int tile = wave * 2 + p, tm = tile >> 2, tn = tile & 3;
    const int srow = tm * 16 + m, ncol = tn * 16 + m;
    v8f acc = {};
#pragma unroll
    for (int k0 = 0; k0 < 64; k0 += 4) {
      const int kk = k0 + kb;
      v2f av, bv;
      av[0] = sW[kk] * sB[kk * PAD + srow];
      av[1] = sW[kk + 1] * sB[(kk + 1) * PAD + srow];
      bv[0] = sX[kk * PAD + ncol];
      bv[1] = sX[(kk + 1) * PAD + ncol];
      acc = wmma_f32(av, bv, acc);
    }
    const size_t sbase = ((size_t)bh * NCHUNK + (size_t)c) * 4096;
#pragma unroll
    for (int r = 0; r < 8; ++r) {
      const int row = tm * 16 + r + half * 8;
      wsS[sbase + row * 64 + ncol] = acc[r];
    }
  }
}

// ---------------------------------------------------------------------------
// Pass 2: exclusive prefix scan over chunk states, scalar decay per chunk.
// Grid 128 blocks x 256 threads; each thread owns one float4 column.
// S_{c+1} loads are independent of the carry -> pipeline ahead of FMA chain.
// ---------------------------------------------------------------------------
__global__ __launch_bounds__(256) void sgr_pass2(const float* __restrict__ wsS,
                                                 const float* __restrict__ wsDc,
                                                 float* __restrict__ wsHp) {
  const int tid = threadIdx.x;
  const int dq = blockIdx.x & 3;
  const int bh = blockIdx.x >> 2;
  const size_t off = (size_t)dq * 1024 + (size_t)tid * 4;
  const float* Sp = wsS + (size_t)bh * NCHUNK * 4096 + off;
  float* Hpp = wsHp + (size_t)bh * NCHUNK * 4096 + off;
  const float* dcp = wsDc + bh * NCHUNK;

  float4 h = {0.0f, 0.0f, 0.0f, 0.0f};
  float4 s_next = *(const float4*)Sp;
  for (int c2 = 0; c2 < NCHUNK; ++c2) {
    const float4 s_cur = s_next;
    if (c2 + 1 < NCHUNK) s_next = *(const float4*)(Sp + (size_t)(c2 + 1) * 4096);
    *(float4*)(Hpp + (size_t)c2 * 4096) = h;  // exclusive prefix
    const float dc = dcp[c2];
    h.x = dc * h.x + s_cur.x;
    h.y = dc * h.y + s_cur.y;
    h.z = dc * h.z + s_cur.z;
    h.w = dc * h.w + s_cur.w;
  }
}

// ---------------------------------------------------------------------------
// Pass 3: fully parallel over (b, h, chunk): Y += diag(e^{L_t}) . C . Hp_c.
// Y is loaded straight into the WMMA C-accumulator.
// ---------------------------------------------------------------------------
#define P3_SMEM_FLOATS (2 * 64 * PAD + 64)

__global__ __launch_bounds__(256) void sgr_pass3(
    const float* __restrict__ Cm, const float* __restrict__ wsHp,
    const float* __restrict__ wsEL, float* __restrict__ Y) {
  extern __shared__ float smem[];
  float* sC = smem;
  float* sHp = sC + 64 * PAD;
  float* sEL = sHp + 64 * PAD;

  const int tid = threadIdx.x;
  const int c = blockIdx.x & (NCHUNK - 1);
  const int bh = blockIdx.x >> 6;
  const int head = bh & (HNUM - 1);
  const int b = bh >> 4;
  const int t0 = c * TCHUNK;

  const int wave = tid >> 5;
  const int lane = tid & 31;
  const int m = lane & 15;
  const int half = lane >> 4;
  const int kb = half * 2;
  const size_t baseBL = (size_t)b * LSEQ;

  {
    const size_t gbase = (baseBL + (size_t)t0) * TSTRIDE + (size_t)head * DDIM;
    const size_t hbase = ((size_t)bh * NCHUNK + (size_t)c) * 4096;
#pragma unroll
    for (int rep = 0; rep < 4; ++rep) {
      const int idx4 = rep * 256 + tid;
      const int r = idx4 >> 4;
      const int j = (idx4 & 15) * 4;
      *(float4*)&sC[r * PAD + j] =
          *(const float4*)&Cm[gbase + (size_t)r * TSTRIDE + (size_t)j];
      *(float4*)&sHp[r * PAD + j] =
          *(const float4*)&wsHp[hbase + (size_t)(r * 64 + j)];
    }
    if (tid < 64) sEL[tid] = wsEL[(size_t)bh * LSEQ + (size_t)(t0 + tid)];
  }
  __syncthreads();

#pragma unroll
  for (int p = 0; p < 2; ++p) {
    const int tile = wave * 2 + p, tm = tile >> 2, tn = tile & 3;
    const int arow = tm * 16 + m, ncol = tn * 16 + m;
    const float el = sEL[arow];
    v8f acc;
#pragma unroll
    for (int r = 0; r < 8; ++r) {  // init accumulator with Y_intra
      const int trow = tm * 16 + r + half * 8;
      acc[r] = Y[(baseBL + (size_t)(t0 + trow)) * TSTRIDE +
                 (size_t)head * DDIM + ncol];
    }
#pragma unroll
    for (int k0 = 0; k0 < 64; k0 += 4) {
      const int kk = k0 + kb;
      v2f av, bv;
      av[0] = el * sC[arow * PAD + kk];
      av[1] = el * sC[arow * PAD + kk + 1];
      bv[0] = sHp[kk * PAD + ncol];
      bv[1] = sHp[(kk + 1) * PAD + ncol];
      acc = wmma_f32(av, bv, acc);
    }
#pragma unroll
    for (int r = 0; r < 8; ++r) {
      const int trow = tm * 16 + r + half * 8;
      Y[(baseBL + (size_t)(t0 + trow)) * TSTRIDE + (size_t)head * DDIM + ncol] =
          acc[r];
    }
  }
}

// ---------------------------------------------------------------------------
// Fallback: monolithic serial-chunk kernel (used if workspace is too small).
// ---------------------------------------------------------------------------
#define MONO_SMEM_FLOATS (5 * 64 * PAD + 5 * 64)

__global__ __launch_bounds__(256) void sgr_mono(
    const float* __restrict__ X, const float* __restrict__ A,
    const float* __restrict__ Bm, const float* __restrict__ Cm,
    const float* __restrict__ l2ab, const float* __restrict__ l2b,
    const float* __restrict__ sema, float* __restrict__ Y) {
  extern __shared__ float smem[];
  float* sX = smem;
  float* sB = sX + 64 * PAD;
  float* sC = sB + 64 * PAD;
  float* sH = sC + 64 * PAD;
  float* sM = sH + 64 * PAD;
  float* sLa = sM + 64 * PAD;
  float* sLc = sLa + 64;
  float* sOma = sLc + 64;
  float* sEL = sOma + 64;
  float* sW = sEL + 64;

  const int tid = threadIdx.x;
  const int head = blockIdx.x & (HNUM - 1);
  const int b = blockIdx.x / HNUM;

  float ab = l2ab[head];
  ab = fminf(fmaxf(ab, -3.32f), -0.015f);
  const float alpha_base = 1.0f - exp2f(ab);
  float btv = l2b[head];
  btv = fminf(fmaxf(btv, -2.0f), 2.0f);
  const float beta = exp2f(btv);
  const float inv_ema = 1.0f / (sema[head] + 1e-6f);

  for (int i = tid; i < 64 * PAD; i += 256) sH[i] = 0.0f;

  const int wave = tid >> 5;
  const int lane = tid & 31;
  const int m = lane & 15;
  const int half = lane >> 4;
  const int kb = half * 2;
  const size_t baseBL = (size_t)b * LSEQ;

  for (int c = 0; c < NCHUNK; ++c) {
    const int t0 = c * TCHUNK;
    __syncthreads();
    {
      const size_t gbase = (baseBL + (size_t)t0) * TSTRIDE + (size_t)head * DDIM;
#pragma unroll
      for (int rep = 0; rep < 4; ++rep) {
        const int idx4 = rep * 256 + tid;
        const int r = idx4 >> 4;
        const int j = (idx4 & 15) * 4;
        const size_t g = gbase + (size_t)r * TSTRIDE + (size_t)j;
        *(float4*)&sX[r * PAD + j] = *(const float4*)&X[g];
        *(float4*)&sB[r * PAD + j] = *(const float4*)&Bm[g];
        *(float4*)&sC[r * PAD + j] = *(const float4*)&Cm[g];
        if (c + 1 < NCHUNK) {
          __builtin_prefetch(&X[g + (size_t)TCHUNK * TSTRIDE], 0, 1);
          __builtin_prefetch(&Bm[g + (size_t)TCHUNK * TSTRIDE], 0, 1);
          __builtin_prefetch(&Cm[g + (size_t)TCHUNK * TSTRIDE], 0, 1);
        }
      }
    }
    __syncthreads();
    {
      const int t = tid >> 2, q = tid & 3;
      float xp = 0.0f, bp = 0.0f;
      const int base = t * PAD + q * 16;
#pragma unroll
      for (int j2 = 0; j2 < 16; ++j2) {
        const float xv = sX[base + j2];
        const float bv = sB[base + j2];
        xp += xv;
        bp += bv * bv;
      }
      xp += __shfl_xor(xp, 1, 4);
      xp += __shfl_xor(xp, 2, 4);
      bp += __shfl_xor(bp, 1, 4);
      bp += __shfl_xor(bp, 2, 4);
      if (q == 0) {
        const float a_log = A[(baseBL + (size_t)(t0 + t)) * HNUM + head];
        const float ad = __expf(a_log);
        const float one_m_ad = 1.0f - ad;
        const float xmean = xp * (1.0f / 64.0f);
        const float pred_err =
            one_m_ad * one_m_ad * xmean * xmean * (bp * (1.0f / 64.0f));
        float boost = fmaxf(tanhf(beta * pred_err * inv_ema), 0.0f);
        float alpha = alpha_base + (1.0f - alpha_base) * boost;
        alpha = fminf(fmaxf(alpha, 0.01f), 0.999f);
        sLa[t] = __logf(alpha);
        sOma[t] = 1.0f - alpha;
      }
    }
    __syncthreads();
    if (tid == 0) {
      float acc = 0.0f;
      for (int i = 0; i < 64; ++i) {
        acc += sLa[i];
        sLc[i] = acc;
      }
    }
    __syncthreads();
    if (tid < 64) {
      const float Lt = sLc[tid];
      const float L63 = sLc[63];
      sEL[tid] = __expf(Lt);
      sW[tid] = __expf(L63 - Lt) * sOma[tid];
    }
    __syncthreads();
#pragma unroll
    for (int p = 0; p < 2; ++p) {
      const int tile = wave * 2 + p, tm = tile >> 2, tn = tile & 3;
      const int arow = tm * 16 + m, brow = tn * 16 + m;
      v8f acc = {};
#pragma unroll
      for (int k0 = 0; k0 < 64; k0 += 4) {
        const int kk = k0 + kb;
        v2f av, bv;
        av[0] = sC[arow * PAD + kk];
        av[1] = sC[arow * PAD + kk + 1];
        bv[0] = sB[brow * PAD + kk];
        bv[1] = sB[brow * PAD + kk + 1];
        acc = wmma_f32(av, bv, acc);
      }
#pragma unroll
      for (int r = 0; r < 8; ++r) {
        const int row = tm * 16 + r + half * 8;
        sM[row * PAD + tn * 16 + m] = acc[r];
      }
    }
    __syncthreads();
    {
      const int t = tid >> 2, q = tid & 3;
      const float Lt = sLc[t];
#pragma unroll
      for (int j2 = 0; j2 < 16; ++j2) {
        const int i = q * 16 + j2;
        float v = 0.0f;
        if (i <= t) v = sM[t * PAD + i] * __expf(Lt - sLc[i]) * sOma[i];
        sM[t * PAD + i] = v;
      }
    }
    __syncthreads();
#pragma unroll
    for (int p = 0; p < 2; ++p) {
      const int tile = wave * 2 + p, tm = tile >> 2, tn = tile & 3;
      const int arow = tm * 16 + m, ncol = tn * 16 + m;
      const float el = sEL[arow];
      v8f acc = {};
#pragma unroll
      for (int k0 = 0; k0 < 64; k0 += 4) {
        const int kk = k0 + kb;
        v2f av, bv;
        av[0] = el * sC[arow * PAD + kk];
        av[1] = el * sC[arow * PAD + kk + 1];
        bv[0] = sH[kk * PAD + ncol];
        bv[1] = sH[(kk + 1) * PAD + ncol];
        acc = wmma_f32(av, bv, acc);
      }
#pragma unroll
      for (int k0 = 0; k0 < 64; k0 += 4) {
        const int kk = k0 + kb;
        v2f av, bv;
        av[0] = sM[arow * PAD + kk];
        av[1] = sM[arow * PAD + kk + 1];
        bv[0] = sX[kk * PAD + ncol];
        bv[1] = sX[(kk + 1) * PAD + ncol];
        acc = wmma_f32(av, bv, acc);
      }
#pragma unroll
      for (int r = 0; r < 8; ++r) {
        const int trow = tm * 16 + r + half * 8;
        Y[(baseBL + (size_t)(t0 + trow)) * TSTRIDE + (size_t)head * DDIM +
          ncol] = acc[r];
      }
    }
    __syncthreads();
#pragma unroll
    for (int p = 0; p < 2; ++p) {
      const int tile = wave * 2 + p, tm = tile >> 2, tn = tile & 3;
      const int srow = tm * 16 + m, ncol = tn * 16 + m;
      v8f acc = {};
#pragma unroll
      for (int k0 = 0; k0 < 64; k0 += 4) {
        const int kk = k0 + kb;
        v2f av, bv;
        av[0] = sW[kk] * sB[kk * PAD + srow];
        av[1] = sW[kk + 1] * sB[(kk + 1) * PAD + srow];
        bv[0] = sX[kk * PAD + ncol];
        bv[1] = sX[(kk + 1) * PAD + ncol];
        acc = wmma_f32(av, bv, acc);
      }
      const float eT = sEL[63];
#pragma unroll
      for (int r = 0; r < 8; ++r) {
        const int row = tm * 16 + r + half * 8;
        const int idx = row * PAD + ncol;
        sH[idx] = eT * sH[idx] + acc[r];
      }
    }
  }
}

extern "C" void kernel_launch(void* const* d_in, const int* in_sizes, int n_in,
                              void* d_out, int out_size, void* d_ws,
                              size_t ws_size, hipStream_t stream) {
  (void)in_sizes; (void)n_in; (void)out_size;
  const float* X = (const float*)d_in[0];
  const float* A = (const float*)d_in[1];
  const float* B = (const float*)d_in[2];
  const float* C = (const float*)d_in[3];
  const float* l2ab = (const float*)d_in[4];
  const float* l2b = (const float*)d_in[5];
  const float* sema = (const float*)d_in[6];
  float* Y = (float*)d_out;

  const size_t elS = (size_t)BSZ * HNUM * NCHUNK * SDIM * DDIM;  // 8.39M
  const size_t elEL = (size_t)BSZ * HNUM * LSEQ;                 // 131072
  const size_t elDc = (size_t)BSZ * HNUM * NCHUNK;               // 2048
  const size_t need = (2 * elS + elEL + elDc) * sizeof(float);

  if (ws_size >= need) {
    float* wsS = (float*)d_ws;
    float* wsHp = wsS + elS;
    float* wsEL = wsHp + elS;
    float* wsDc = wsEL + elEL;
    sgr_pass1<<<dim3(BSZ * HNUM * NCHUNK), dim3(256),
                P1_SMEM_FLOATS * sizeof(float), stream>>>(
        X, A, B, C, l2ab, l2b, sema, Y, wsS, wsEL, wsDc);
    sgr_pass2<<<dim3(BSZ * HNUM * 4), dim3(256), 0, stream>>>(wsS, wsDc, wsHp);
    sgr_pass3<<<dim3(BSZ * HNUM * NCHUNK), dim3(256),
                P3_SMEM_FLOATS * sizeof(float), stream>>>(C, wsHp, wsEL, Y);
  } else {
    sgr_mono<<<dim3(BSZ * HNUM), dim3(256), MONO_SMEM_FLOATS * sizeof(float),
               stream>>>(X, A, B, C, l2ab, l2b, sema, Y);
  }
}